// myYOLOv3_76364518523559
// MI455X (gfx1250) — compile-verified
//
#include <hip/hip_runtime.h>
#include <stdint.h>

#define N_CELLS   63072
#define N_PRED    (N_CELLS * 3)   // 189216 candidates per batch
#define PRED_CH   27
#define TOPK      512
#define SORT_CHUNK 2048
#define CONF_TH   0.02f
#define NMS_TH    0.5f

typedef float v2f __attribute__((ext_vector_type(2)));
typedef float v8f __attribute__((ext_vector_type(8)));

#if defined(__HIP_DEVICE_COMPILE__) && defined(__gfx1250__)
#define IS_GFX1250 1
#else
#define IS_GFX1250 0
#endif

#if IS_GFX1250 && __has_builtin(__builtin_amdgcn_wmma_f32_16x16x4_f32)
#define USE_WMMA 1
#else
#define USE_WMMA 0
#endif

__device__ __constant__ float c_anch[3][3][3] = {
    {{10.f, 13.f, 8.f},   {16.f, 30.f, 12.f},  {33.f, 23.f, 16.f}},
    {{30.f, 61.f, 24.f},  {62.f, 45.f, 32.f},  {59.f, 119.f, 48.f}},
    {{116.f, 90.f, 64.f}, {156.f, 198.f, 96.f},{373.f, 326.f, 128.f}}};

static __device__ __forceinline__ float sigm(float x) {
  return 1.0f / (1.0f + expf(-x));
}

// ---------------- decode: boxes + scores, zero keep ----------------
__global__ void k_decode(const float* __restrict__ pred, float* __restrict__ boxes,
                         float* __restrict__ scores, float* __restrict__ keep,
                         int total) {
  int i = blockIdx.x * blockDim.x + threadIdx.x;
  if (i >= total) return;
  int n = i % N_PRED;
  int c = n / 3, a = n % 3;
  int lvl, cl, hs, ds, st;
  if (c < 55296)      { lvl = 0; cl = c;          hs = 48; ds = 24; st = 8;  }
  else if (c < 62208) { lvl = 1; cl = c - 55296;  hs = 24; ds = 12; st = 16; }
  else                { lvl = 2; cl = c - 62208;  hs = 12; ds = 6;  st = 32; }
  float gx = (float)(cl / (hs * ds));
  float gy = (float)((cl / ds) % hs);
  float gz = (float)(cl % ds);
  const float* p = pred + (size_t)i * PRED_CH;
  float fs = (float)st;
  float cx = (sigm(p[1]) + gx) * fs;
  float cy = (sigm(p[2]) + gy) * fs;
  float cz = (sigm(p[3]) + gz) * fs;
  float wx = expf(p[4]) * c_anch[lvl][a][0];
  float wy = expf(p[5]) * c_anch[lvl][a][1];
  float wz = expf(p[6]) * c_anch[lvl][a][2];
  float* bo = boxes + (size_t)i * 6;
  bo[0] = cx - wx * 0.5f; bo[1] = cy - wy * 0.5f; bo[2] = cz - wz * 0.5f;
  bo[3] = cx + wx * 0.5f; bo[4] = cy + wy * 0.5f; bo[5] = cz + wz * 0.5f;
  // softmax over 20 classes, take class 0
  float m = p[7];
#pragma unroll
  for (int k = 8; k < 27; ++k) m = fmaxf(m, p[k]);
  float sum = 0.f;
#pragma unroll
  for (int k = 7; k < 27; ++k) sum += expf(p[k] - m);
  float cls0 = expf(p[7] - m) / sum;
  scores[i] = sigm(p[0]) * cls0;
  keep[i] = 0.0f;   // keep mask default: false
}

// ---------------- pack sortable 64-bit keys ----------------
__global__ void k_keys(const float* __restrict__ scoresB,
                       unsigned long long* __restrict__ keys) {
  int t = blockIdx.x * blockDim.x + threadIdx.x;
  if (t >= N_PRED) return;
  float sc = scoresB[t];
  float s = (sc >= CONF_TH) ? sc : -1.0f;
  unsigned int bits = __float_as_uint(s);
  unsigned int u = (bits & 0x80000000u) ? ~bits : (bits | 0x80000000u);
  // descending score, ties -> lower index first (matches lax.top_k)
  keys[t] = ((unsigned long long)u << 32) |
            (unsigned long long)(~(unsigned int)t);
}

// ---------------- bitonic "top-512 of 2048" pass ----------------
__global__ __launch_bounds__(1024)
void k_sort(const unsigned long long* __restrict__ in,
            unsigned long long* __restrict__ out, int n_in) {
  __shared__ unsigned long long sm[SORT_CHUNK];
  const int tid = threadIdx.x;
  const long long base = (long long)blockIdx.x * SORT_CHUNK;

#if IS_GFX1250
  // CDNA5 async copy: global -> LDS, tracked by ASYNCcnt.
  // LDS byte offset == low 32 bits of the flat address (LDS aperture rule).
  for (int e = tid; e < SORT_CHUNK; e += 1024) {
    long long g = base + e;
    long long gc = (g < (long long)n_in - 1) ? g : (long long)n_in - 1;
    unsigned int lds_off = (unsigned int)(uintptr_t)(&sm[e]);
    unsigned long long gaddr = (unsigned long long)(uintptr_t)(in + gc);
    asm volatile("global_load_async_to_lds_b64 %0, %1, off"
                 :: "v"(lds_off), "v"(gaddr) : "memory");
  }
  asm volatile("s_wait_asynccnt 0" ::: "memory");
  __syncthreads();
  for (int e = tid; e < SORT_CHUNK; e += 1024) {
    if (base + e >= n_in) sm[e] = 0ull;   // pad sorts to the bottom
  }
  __syncthreads();
#else
  for (int e = tid; e < SORT_CHUNK; e += 1024) {
    long long g = base + e;
    sm[e] = (g < n_in) ? in[g] : 0ull;
  }
  __syncthreads();
#endif

  // bitonic sort, descending
  for (int k = 2; k <= SORT_CHUNK; k <<= 1) {
    for (int j = k >> 1; j > 0; j >>= 1) {
      for (int e = tid; e < SORT_CHUNK; e += 1024) {
        int l = e ^ j;
        if (l > e) {
          unsigned long long A = sm[e], Bv = sm[l];
          bool up = ((e & k) == 0);
          if (up ? (A < Bv) : (A > Bv)) { sm[e] = Bv; sm[l] = A; }
        }
      }
      __syncthreads();
    }
  }
  if (tid < TOPK) out[(size_t)blockIdx.x * TOPK + tid] = sm[tid];
}

// ---------------- gather top-512 boxes, areas, scores, indices ----------------
__global__ void k_gather(const unsigned long long* __restrict__ keys,
                         const float* __restrict__ boxesB,
                         float* __restrict__ b_top, float* __restrict__ areaT,
                         float* __restrict__ topS, unsigned int* __restrict__ topIdx) {
  int t = threadIdx.x;
  unsigned long long key = keys[t];
  unsigned int u = (unsigned int)(key >> 32);
  unsigned int bits = (u & 0x80000000u) ? (u ^ 0x80000000u) : ~u;
  float s = __uint_as_float(bits);
  unsigned int idx = ~(unsigned int)(key & 0xFFFFFFFFu);
  const float* bb = boxesB + (size_t)idx * 6;
  float b0 = bb[0], b1 = bb[1], b2 = bb[2], b3 = bb[3], b4 = bb[4], b5 = bb[5];
  float* o = b_top + t * 6;
  o[0] = b0; o[1] = b1; o[2] = b2; o[3] = b3; o[4] = b4; o[5] = b5;
  areaT[t] = (b3 - b0) * (b4 - b1) * (b5 - b2);
  topS[t] = s;
  topIdx[t] = idx;
}

// ---------------- 512x512 IoU matrix in 16x16 WMMA tiles ----------------
// D = A x B + C with A = [area_i | 1] (16x4, K=2..3 zero), B = [1 ; area_j],
// C = -inter  =>  D[i][j] = area_i + area_j - inter ; iou = inter / D.
__global__ void k_iou(const float* __restrict__ b_top,
                      const float* __restrict__ areaT,
                      float* __restrict__ iou) {
  const int lane = threadIdx.x;                 // one wave32 per tile
  const int i0 = blockIdx.y * 16, j0 = blockIdx.x * 16;
  const int jn = j0 + (lane & 15);
  const int ibase = i0 + ((lane >> 4) << 3);    // f32 C/D layout: M = r (+8 for hi lanes)
  float bj[6];
#pragma unroll
  for (int k = 0; k < 6; ++k) bj[k] = b_top[jn * 6 + k];
  float interArr[8];
  v8f cacc;
#pragma unroll
  for (int r = 0; r < 8; ++r) {
    const float* bi = b_top + (ibase + r) * 6;
    float d0 = fmaxf(fminf(bi[3], bj[3]) - fmaxf(bi[0], bj[0]), 1e-28f);
    float d1 = fmaxf(fminf(bi[4], bj[4]) - fmaxf(bi[1], bj[1]), 1e-28f);
    float d2 = fmaxf(fminf(bi[5], bj[5]) - fmaxf(bi[2], bj[2]), 1e-28f);
    float inter = d0 * d1 * d2;
    interArr[r] = inter;
    cacc[r] = -inter;
  }
  v8f d;
#if USE_WMMA
  v2f av, bv;
  // A 16x4 layout: lanes 0-15 = rows M, vgpr0=K0, vgpr1=K1; lanes 16-31 = K2,K3 (zero)
  av.x = (lane < 16) ? areaT[i0 + lane] : 0.f;
  av.y = (lane < 16) ? 1.f : 0.f;
  // B 4x16 layout: lanes 0-15 = cols N, vgpr0=K0 row (ones), vgpr1=K1 row (area_j)
  bv.x = (lane < 16) ? 1.f : 0.f;
  bv.y = (lane < 16) ? areaT[j0 + lane] : 0.f;
  d = __builtin_amdgcn_wmma_f32_16x16x4_f32(false, av, false, bv,
                                            (short)0, cacc, false, false);
#else
#pragma unroll
  for (int r = 0; r < 8; ++r) d[r] = areaT[ibase + r] + areaT[jn] + cacc[r];
#endif
#pragma unroll
  for (int r = 0; r < 8; ++r)
    iou[(size_t)(ibase + r) * TOPK + jn] = interArr[r] / d[r];
}

// ---------------- sequential greedy NMS + scatter ----------------
__global__ __launch_bounds__(TOPK)
void k_nms(const float* __restrict__ iou, const float* __restrict__ topS,
           const unsigned int* __restrict__ topIdx, float* __restrict__ keepB) {
  __shared__ int keep[TOPK];
  __shared__ float ss[TOPK];
  int t = threadIdx.x;
  keep[t] = 1;
  ss[t] = topS[t];
  __syncthreads();
  for (int i = 0; i < TOPK; ++i) {
    bool alive = (keep[i] != 0) && (ss[i] > 0.0f);
    if (alive && t > i && iou[(size_t)i * TOPK + t] > NMS_TH) keep[t] = 0;
    __syncthreads();
  }
  if (keep[t] && ss[t] > 0.0f) keepB[topIdx[t]] = 1.0f;
}

// ---------------- host launcher ----------------
extern "C" void kernel_launch(void* const* d_in, const int* in_sizes, int n_in,
                              void* d_out, int out_size, void* d_ws, size_t ws_size,
                              hipStream_t stream) {
  (void)n_in; (void)out_size; (void)ws_size;
  const float* pred = (const float*)d_in[0];
  int B = in_sizes[0] / (N_PRED * PRED_CH);
  if (B < 1) B = 1;

  float* out = (float*)d_out;
  float* boxes  = out;                                   // [B, N, 6]
  float* scores = out + (size_t)B * N_PRED * 6;          // [B, N]
  float* keep   = out + (size_t)B * N_PRED * 7;          // [B, N]

  char* w = (char*)d_ws;
  unsigned long long* keysA = (unsigned long long*)w;                       // N_PRED keys
  unsigned long long* keysB = (unsigned long long*)(w + (size_t)N_PRED * 8);// 47616 keys
  char* w2 = w + (size_t)N_PRED * 8 + 47616ull * 8;
  float* b_top = (float*)w2;                              // 512*6
  float* areaT = (float*)(w2 + 512 * 6 * 4);              // 512
  float* topS  = (float*)(w2 + 512 * 6 * 4 + 512 * 4);    // 512
  unsigned int* topIdx = (unsigned int*)(w2 + 512 * 6 * 4 + 2 * 512 * 4);
  float* iouM  = (float*)(w2 + 512 * 6 * 4 + 3 * 512 * 4); // 512*512

  int total = B * N_PRED;
  k_decode<<<(total + 255) / 256, 256, 0, stream>>>(pred, boxes, scores, keep, total);

  for (int b = 0; b < B; ++b) {
    k_keys<<<(N_PRED + 255) / 256, 256, 0, stream>>>(scores + (size_t)b * N_PRED, keysA);
    const unsigned long long* src = keysA;
    unsigned long long* dst = keysB;
    int n = N_PRED;
    for (;;) {   // 189216 -> 47616 -> 12288 -> 3072 -> 1024 -> 512 (fixed)
      int blocks = (n + SORT_CHUNK - 1) / SORT_CHUNK;
      k_sort<<<blocks, 1024, 0, stream>>>(src, dst, n);
      if (blocks == 1) break;
      n = blocks * TOPK;
      const unsigned long long* tmp = dst;
      dst = (unsigned long long*)src;
      src = tmp;
    }
    k_gather<<<1, TOPK, 0, stream>>>(dst, boxes + (size_t)b * N_PRED * 6,
                                     b_top, areaT, topS, topIdx);
    k_iou<<<dim3(32, 32, 1), dim3(32, 1, 1), 0, stream>>>(b_top, areaT, iouM);
    k_nms<<<1, TOPK, 0, stream>>>(iouM, topS, topIdx, keep + (size_t)b * N_PRED);
  }
}